// Compressor_85538568667345
// MI455X (gfx1250) — compile-verified
//
#include <hip/hip_runtime.h>
#include <stdint.h>

// CDNA5 / gfx1250, wave32. Blockwise 8x8 DCT + int8 binning of an 8192x8192 f32 matrix.
//
// A 16x16 tile of x holds a 2x2 group of 8x8 DCT blocks. With the block-diagonal
// matrix D = diag(M, M) (M = 8x8 orthonormal DCT-II, M[element][freq]),
// D^T * T * D computes all four per-block DCTs at once; each 16x16x16 product is
// 4 chained V_WMMA_F32_16X16X4_F32 ops (K split 4x4). Memory-bound (~528 MB
// moved => ~23 us at 23.3 TB/s): single pass over x, 1KB LDS round-trip for the
// inter-pass operand relayout, coalesced b128 output stores.

typedef __attribute__((ext_vector_type(2))) float v2f;
typedef __attribute__((ext_vector_type(8))) float v8f;

#define MAT_N    8192
#define NB       (MAT_N / 8)          // 1024 DCT blocks per row/col
#define TILES    (MAT_N / 16)         // 512 tiles per row/col
#define NTILES   (TILES * TILES)      // 262144
#define TPW      4                    // tiles per wave
#define WAVES_PB 8
#define THREADS  (WAVES_PB * 32)
#define NBLOCKS  (NTILES / (TPW * WAVES_PB))  // 8192
#define S        20                   // LDS row stride (floats), pads banks

// D = diag(M, M); M[e][g] = dct[e*8+g]
__device__ __forceinline__ float dval(const float* __restrict__ dct, int k, int m) {
  return ((k < 8) == (m < 8)) ? dct[(k & 7) * 8 + (m & 7)] : 0.0f;
}

// max across each 8-lane group via constant-pattern ds_swizzle (no index VGPRs):
// group-of-32 mode, offset = xor_mask<<10 | and_mask(0x1f)
__device__ __forceinline__ float grp8_max(float m) {
  m = fmaxf(m, __int_as_float(__builtin_amdgcn_ds_swizzle(__float_as_int(m), 0x041F)));
  m = fmaxf(m, __int_as_float(__builtin_amdgcn_ds_swizzle(__float_as_int(m), 0x081F)));
  m = fmaxf(m, __int_as_float(__builtin_amdgcn_ds_swizzle(__float_as_int(m), 0x101F)));
  return m;
}

__global__ __launch_bounds__(THREADS) void dct8_bin_kernel(
    const float* __restrict__ x, const float* __restrict__ dct,
    float* __restrict__ out_idx, float* __restrict__ out_big) {
  __shared__ float lds[WAVES_PB][16 * S];
  const int  lane = threadIdx.x & 31;
  const int  wave = threadIdx.x >> 5;
  const bool hi   = lane >= 16;        // lanes 16-31 carry K+2/K+3 (A) and rows+2/+3 (B)
  const int  l    = lane & 15;
  float* sl = lds[wave];

  // Constant operand: A-layout of D^T (pass 1) == B-layout of D (pass 2).
  v2f dmat[4];
#pragma unroll
  for (int j = 0; j < 4; ++j) {
    const int k0 = 4 * j + (hi ? 2 : 0);
    dmat[j].x = dval(dct, k0,     l);
    dmat[j].y = dval(dct, k0 + 1, l);
  }

  const int waveGlobal = blockIdx.x * WAVES_PB + wave;
#pragma unroll 1
  for (int t = 0; t < TPW; ++t) {
    const int ti = (waveGlobal * TPW + t) >> 9;          // tile row
    const int tj = (waveGlobal * TPW + t) & (TILES - 1); // tile col
    const float* xp = x + (size_t)ti * 16 * MAT_N + (size_t)tj * 16 + l;

    // Load X tile directly in B-layout: chunk j holds rows 4j..4j+3.
    v2f bx[4];
#pragma unroll
    for (int j = 0; j < 4; ++j) {
      const int r0 = 4 * j + (hi ? 2 : 0);
      bx[j].x = xp[(size_t)r0 * MAT_N];
      bx[j].y = xp[(size_t)(r0 + 1) * MAT_N];
    }

    // Pass 1: P = D^T * X   (4x V_WMMA_F32_16X16X4_F32, K accumulated)
    v8f p = {};
#pragma unroll
    for (int j = 0; j < 4; ++j)
      p = __builtin_amdgcn_wmma_f32_16x16x4_f32(false, dmat[j], false, bx[j],
                                                (short)0, p, false, false);

    // Stage P (C/D layout) row-major into LDS, re-read in A-layout.
#pragma unroll
    for (int r = 0; r < 8; ++r)
      sl[(r + (hi ? 8 : 0)) * S + l] = p[r];
    asm volatile("s_wait_dscnt 0x0" ::: "memory");

    v2f ap[4];
#pragma unroll
    for (int j = 0; j < 4; ++j) {
      const int b = l * S + 4 * j + (hi ? 2 : 0);
      ap[j].x = sl[b];
      ap[j].y = sl[b + 1];
    }

    // Pass 2: C = P * D
    v8f c = {};
#pragma unroll
    for (int j = 0; j < 4; ++j)
      c = __builtin_amdgcn_wmma_f32_16x16x4_f32(false, ap[j], false, dmat[j],
                                                (short)0, c, false, false);

    // Each 8-lane group owns one 8x8 block (8 VGPRs x 8 lanes = 64 coeffs).
    float m = 0.0f;
#pragma unroll
    for (int r = 0; r < 8; ++r) m = fmaxf(m, fabsf(c[r]));
    m = grp8_max(m);
    // 1-ulp reciprocal is fine for a 127-level quantizer; avoids the IEEE
    // divide expansion (v_div_scale/v_div_fmas chain + trans hazard nop).
    const float scale = 127.0f * __builtin_amdgcn_rcpf(m);

    // Quantize (round-half-even like jnp.round). |c*scale| <= 127 by
    // construction, so the int8 round-trip is the identity on rintf's result.
    const int I = hi ? 1 : 0;           // row block in tile
    const int J = (l >> 3) & 1;         // col block in tile
    const int h = lane & 7;             // col within block
#pragma unroll
    for (int r = 0; r < 8; ++r)
      sl[(I * 2 + J) * 64 + r * 8 + h] = rintf(c[r] * scale);
    asm volatile("s_wait_dscnt 0x0" ::: "memory");

    // Drain the repacked tile: 32B per lane, 128b stores, blocks (b1,2tj) and
    // (b1,2tj+1) are contiguous so each 16-lane half writes a 512B burst.
    {
      const int    B  = lane >> 3;        // which of the 4 blocks this lane drains
      const int    o  = (lane & 7) * 8;   // float offset within the block's 64
      const size_t b1 = 2 * (size_t)ti + (B >> 1);
      const size_t b2 = 2 * (size_t)tj + (B & 1);
      float* dst = out_idx + (b1 * NB + b2) * 64 + o;
      const float4 v0 = *(const float4*)(sl + B * 64 + o);
      const float4 v1 = *(const float4*)(sl + B * 64 + o + 4);
      *(float4*)dst       = v0;
      *(float4*)(dst + 4) = v1;
      if ((lane & 7) == 0) out_big[b1 * NB + b2] = m;
    }
    // Keep this tile's LDS drain ordered before next tile's staging stores.
    asm volatile("s_wait_dscnt 0x0" ::: "memory");
  }
}

extern "C" void kernel_launch(void* const* d_in, const int* in_sizes, int n_in,
                              void* d_out, int out_size, void* d_ws, size_t ws_size,
                              hipStream_t stream) {
  (void)in_sizes; (void)n_in; (void)out_size; (void)d_ws; (void)ws_size;
  const float* x   = (const float*)d_in[0];   // (8192, 8192) f32
  const float* dct = (const float*)d_in[1];   // (8, 8) f32, M[element][freq]
  float* out_idx = (float*)d_out;                              // 1024*1024*64 bin values
  float* out_big = out_idx + (size_t)NB * NB * 64;             // 1024*1024 block maxima
  hipLaunchKernelGGL(dct8_bin_kernel, dim3(NBLOCKS), dim3(THREADS), 0, stream,
                     x, dct, out_idx, out_big);
}